// GCN_11450382811785
// MI455X (gfx1250) — compile-verified
//
#include <hip/hip_runtime.h>

typedef __attribute__((ext_vector_type(2))) float v2f;
typedef __attribute__((ext_vector_type(8))) float v8f;

#define IN_F 64
#define HID_F 128
#define OUT_F 40

// ---------------- CSR construction ----------------

__global__ void gcn_zero_int(int* __restrict__ p, int n) {
  int i = blockIdx.x * blockDim.x + threadIdx.x;
  if (i < n) p[i] = 0;
}

__global__ void gcn_hist(const int* __restrict__ src, int* __restrict__ deg, int E) {
  int e = blockIdx.x * blockDim.x + threadIdx.x;
  if (e < E) atomicAdd(&deg[src[e]], 1);
}

// single-block exclusive scan of deg -> row_ptr (and cursor copy), n up to a few 100k
__global__ void gcn_scan(const int* __restrict__ deg, int* __restrict__ row_ptr,
                         int* __restrict__ cursor, int n) {
  __shared__ int sdata[1024];
  __shared__ int carry;
  const int t = threadIdx.x;
  if (t == 0) carry = 0;
  __syncthreads();
  for (int base = 0; base < n; base += 1024) {
    const int i = base + t;
    const int v = (i < n) ? deg[i] : 0;
    sdata[t] = v;
    __syncthreads();
    // Hillis-Steele inclusive scan in LDS
    for (int off = 1; off < 1024; off <<= 1) {
      int tv = (t >= off) ? sdata[t - off] : 0;
      __syncthreads();
      sdata[t] += tv;
      __syncthreads();
    }
    const int excl = sdata[t] - v;
    const int c = carry;           // read before update barrier
    if (i < n) {
      row_ptr[i] = c + excl;
      cursor[i]  = c + excl;
    }
    __syncthreads();
    if (t == 0) carry += sdata[1023];
    __syncthreads();
  }
  if (t == 0) row_ptr[n] = carry;
}

__global__ void gcn_scatter(const int* __restrict__ src, const int* __restrict__ dst,
                            const float* __restrict__ vals, int* __restrict__ cursor,
                            int* __restrict__ csr_col, float* __restrict__ csr_val, int E) {
  int e = blockIdx.x * blockDim.x + threadIdx.x;
  if (e < E) {
    const int r = src[e];
    const int p = atomicAdd(&cursor[r], 1);
    csr_col[p] = dst[e];
    csr_val[p] = vals[e];
  }
}

// ---------------- SpMM gather: one wave32 per row ----------------
// y[r, :] = sum_{i in [rowptr[r], rowptr[r+1])} csr_val[i] * x[csr_col[i], :]
template <int F>
__global__ void gcn_spmm_csr(const int* __restrict__ rowptr, const int* __restrict__ col,
                             const float* __restrict__ val, const float* __restrict__ x,
                             float* __restrict__ y, int n) {
  const int row  = (blockIdx.x * blockDim.x + threadIdx.x) >> 5;
  const int lane = threadIdx.x & 31;
  if (row >= n) return;
  const int s = rowptr[row];
  const int e = rowptr[row + 1];
  if constexpr (F == 64) {
    float2 acc = make_float2(0.f, 0.f);
    const float* xb = x + (size_t)lane * 2;
    for (int i = s; i < e; ++i) {
      const int   c = col[i];    // wave-uniform -> scalar load
      const float v = val[i];
      const float2 xv = *(const float2*)(xb + (size_t)c * 64);
      acc.x += v * xv.x;
      acc.y += v * xv.y;
    }
    *(float2*)(y + (size_t)row * 64 + lane * 2) = acc;
  } else {
    float4 acc = make_float4(0.f, 0.f, 0.f, 0.f);
    const float* xb = x + (size_t)lane * 4;
    for (int i = s; i < e; ++i) {
      const int   c = col[i];
      const float v = val[i];
      const float4 xv = *(const float4*)(xb + (size_t)c * 128);
      acc.x += v * xv.x;
      acc.y += v * xv.y;
      acc.z += v * xv.z;
      acc.w += v * xv.w;
    }
    *(float4*)(y + (size_t)row * 128 + lane * 4) = acc;
  }
}

// ---------------- GEMM1: H = relu(AX[ N x 64 ] * W1[ 64 x 128 ]) ----------------
// block = 256 threads = 8 waves; block covers 16 rows x 128 cols; wave w -> 16-col tile
__global__ void gcn_gemm1_relu(const float* __restrict__ A, const float* __restrict__ W,
                               float* __restrict__ H, int N) {
  const int row0 = blockIdx.x * 16;
  const int wave = threadIdx.x >> 5;
  const int lane = threadIdx.x & 31;
  const int col0 = wave * 16;
  const int m  = lane & 15;          // A row within tile / also B col index
  const int kh = (lane >> 4) * 2;    // K sub-offset for A/B fragments

  const int arow = min(row0 + m, N - 1);
  const float* ap = A + (size_t)arow * IN_F + kh;
  const float* bp = W + (size_t)kh * HID_F + col0 + m;

  v8f c = {};
#pragma unroll
  for (int k = 0; k < IN_F; k += 4) {
    v2f a = *(const v2f*)(ap + k);
    v2f b;
    b.x = bp[(size_t)k * HID_F];
    b.y = bp[(size_t)(k + 1) * HID_F];
    c = __builtin_amdgcn_wmma_f32_16x16x4_f32(false, a, false, b, (short)0, c,
                                              false, false);
  }

  const int mbase = 8 * (lane >> 4);
#pragma unroll
  for (int i = 0; i < 8; ++i) {
    const int r = row0 + mbase + i;
    if (r < N) {
      float v = c[i];
      H[(size_t)r * HID_F + col0 + m] = v > 0.f ? v : 0.f;
    }
  }
}

// ---------------- GEMM2: OUT = AH[ N x 128 ] * W2[ 128 x 40 ] ----------------
// block = 96 threads = 3 waves; 3 x 16 = 48 cols cover the 40 output cols
__global__ void gcn_gemm2(const float* __restrict__ A, const float* __restrict__ W,
                          float* __restrict__ O, int N) {
  const int row0 = blockIdx.x * 16;
  const int wave = threadIdx.x >> 5;
  const int lane = threadIdx.x & 31;
  const int col0 = wave * 16;
  const int m  = lane & 15;
  const int kh = (lane >> 4) * 2;
  const int nc = col0 + m;                 // global output column for B/D
  const int ncc = min(nc, OUT_F - 1);      // clamp address; cols >= 40 never stored

  const int arow = min(row0 + m, N - 1);
  const float* ap = A + (size_t)arow * HID_F + kh;

  v8f c = {};
#pragma unroll
  for (int k = 0; k < HID_F; k += 4) {
    v2f a = *(const v2f*)(ap + k);
    v2f b;
    const int kk = k + kh;
    b.x = W[(size_t)kk * OUT_F + ncc];
    b.y = W[(size_t)(kk + 1) * OUT_F + ncc];
    c = __builtin_amdgcn_wmma_f32_16x16x4_f32(false, a, false, b, (short)0, c,
                                              false, false);
  }

  if (nc < OUT_F) {
    const int mbase = 8 * (lane >> 4);
#pragma unroll
    for (int i = 0; i < 8; ++i) {
      const int r = row0 + mbase + i;
      if (r < N) O[(size_t)r * OUT_F + nc] = c[i];
    }
  }
}

// ---------------- host launcher ----------------

extern "C" void kernel_launch(void* const* d_in, const int* in_sizes, int n_in,
                              void* d_out, int out_size, void* d_ws, size_t ws_size,
                              hipStream_t stream) {
  const int*   src  = (const int*)d_in[0];
  const int*   dst  = (const int*)d_in[1];
  const float* vals = (const float*)d_in[2];
  const float* x    = (const float*)d_in[3];
  const float* W1   = (const float*)d_in[4];
  const float* W2   = (const float*)d_in[5];
  float* out        = (float*)d_out;

  const int E = in_sizes[0];
  const int N = in_sizes[3] / IN_F;

  // workspace carve-out (256B aligned slices)
  size_t off = 0;
  auto take = [&](size_t bytes) {
    size_t o = off;
    off = (off + bytes + 255) & ~(size_t)255;
    return o;
  };
  char* ws = (char*)d_ws;
  int*   deg     = (int*)(ws + take((size_t)N * 4));
  int*   row_ptr = (int*)(ws + take((size_t)(N + 1) * 4));
  int*   cursor  = (int*)(ws + take((size_t)N * 4));
  int*   csr_col = (int*)(ws + take((size_t)E * 4));
  float* csr_val = (float*)(ws + take((size_t)E * 4));
  float* AX      = (float*)(ws + take((size_t)N * IN_F * 4));
  float* H1      = (float*)(ws + take((size_t)N * HID_F * 4));
  float* AH1     = (float*)(ws + take((size_t)N * HID_F * 4));
  (void)ws_size;

  // 1) CSR build
  gcn_zero_int<<<(N + 255) / 256, 256, 0, stream>>>(deg, N);
  gcn_hist<<<(E + 255) / 256, 256, 0, stream>>>(src, deg, E);
  gcn_scan<<<1, 1024, 0, stream>>>(deg, row_ptr, cursor, N);
  gcn_scatter<<<(E + 255) / 256, 256, 0, stream>>>(src, dst, vals, cursor,
                                                   csr_col, csr_val, E);

  const int spmm_blocks = (N + 7) / 8;  // 8 waves (rows) per 256-thread block
  const int gemm_rows   = (N + 15) / 16;

  // 2) AX = A @ x
  gcn_spmm_csr<IN_F><<<spmm_blocks, 256, 0, stream>>>(row_ptr, csr_col, csr_val,
                                                      x, AX, N);
  // 3) H1 = relu(AX @ W1)
  gcn_gemm1_relu<<<gemm_rows, 256, 0, stream>>>(AX, W1, H1, N);
  // 4) AH1 = A @ H1
  gcn_spmm_csr<HID_F><<<spmm_blocks, 256, 0, stream>>>(row_ptr, csr_col, csr_val,
                                                       H1, AH1, N);
  // 5) out = AH1 @ W2
  gcn_gemm2<<<gemm_rows, 96, 0, stream>>>(AH1, W2, out, N);
}